// SAN2d_74079595922027
// MI455X (gfx1250) — compile-verified
//
#include <hip/hip_runtime.h>
#include <hip/hip_bf16.h>
#include <math.h>

// ---------------------------------------------------------------------------
// SAN2d on MI455X (gfx1250, wave32):
//   per filter w in {3x3,5x5,7x7}:
//     sim  = conv_same(x, w)                       [WMMA f32 16x16x4 implicit GEMM,
//                                                   async global->LDS staging]
//     th_b = 2621-th largest |sim_b|               [exact 4-pass radix select]
//     out += conv_same(mask(sim, th), w)           [same WMMA kernel, masked+accum]
// ---------------------------------------------------------------------------

typedef __attribute__((ext_vector_type(2))) float v2f;
typedef __attribute__((ext_vector_type(8))) float v8f;

typedef __attribute__((address_space(1))) int g_i32;   // global b32 cell
typedef __attribute__((address_space(3))) int l_i32;   // LDS b32 cell

#if defined(__has_builtin)
#if __has_builtin(__builtin_amdgcn_wmma_f32_16x16x4_f32)
#define SAN_USE_WMMA 1
#endif
#if __has_builtin(__builtin_amdgcn_global_load_async_to_lds_b32) && \
    __has_builtin(__builtin_amdgcn_s_wait_asynccnt)
#define SAN_USE_ASYNC 1
#endif
#endif

#define IMG_H 512
#define IMG_W 512
#define IMG_N (IMG_H * IMG_W)       // 262144
#define BATCH 64
#define TOPK 2621

// ---------------------------------------------------------------------------
// Conv kernel: 4 waves / block, each wave computes one 16x16 output tile via
// banded-Toeplitz WMMA:  D[m][n] = sum_{dy,c} A_dy,c[m][k] * B_dy,c[k][n]
//   A[m][k] = patch(y0+m+dy-R, x0-R + 4c+k)              (LDS-staged)
//   B[k][n] = w[dy][4c+k-n]  if 0 <= 4c+k-n < L else 0   (band matrix)
// WMMA f32 16x16x4 operand mapping (wave32, ISA 7.12.2):
//   A (16x4): lane = {M=lane&15}, VGPRs hold K = 2*(lane>>4) + {0,1}
//   B (4x16): lane = {N=lane&15}, VGPRs hold K = 2*(lane>>4) + {0,1}  (mirrors A)
//   C/D (16x16): vgpr v, lane l -> row = v + 8*(l>>4), col = l&15
// ---------------------------------------------------------------------------
template <int L, bool MASKED_ACCUM>
__global__ __launch_bounds__(128) void conv_wmma_kernel(
    const float* __restrict__ in, const float* __restrict__ w,
    const float* __restrict__ th, float* __restrict__ out) {
  constexpr int R  = L / 2;
  constexpr int PR = 16 + 2 * R;              // patch rows
  constexpr int PW = 16 + 2 * R;              // valid patch cols
  constexpr int LW = 26;                      // LDS row stride (floats): pads K-chunk
                                              // overreach + conflict-free b64 reads
  constexpr int NC = (16 + L - 1 + 3) / 4;    // number of K=4 chunks

  __shared__ float patch[4][PR * LW];
  __shared__ float wsh[L * L];

  const int tid  = threadIdx.x;
  const int wave = tid >> 5;                  // wave32
  const int lane = tid & 31;

  for (int i = tid; i < L * L; i += 128) wsh[i] = w[i];

  const int g  = blockIdx.x * 4 + wave;       // tile id: 64 * 32 * 32 tiles
  const int b  = g >> 10;
  const int ty = (g >> 5) & 31;
  const int tx = g & 31;
  const int y0 = ty * 16, x0 = tx * 16;

  const float* src = in + (size_t)b * IMG_N;
  float thv = 0.f;
  if (MASKED_ACCUM) thv = th[b];

  // Stage (zero-padded) input patch into LDS. Cols >= PW are zeroed so
  // band-overreach multiplies 0*0. Masked pass applies the top-k threshold
  // in-register while staging.
  float* P = patch[wave];

  auto stage_sync = [&]() {
    for (int i = lane; i < PR * LW; i += 32) {
      const int row = i / LW, col = i % LW;
      const int gy = y0 - R + row, gx = x0 - R + col;
      float v = 0.f;
      if (col < PW && gy >= 0 && gy < IMG_H && gx >= 0 && gx < IMG_W) {
        v = src[gy * IMG_W + gx];
        if (MASKED_ACCUM) v = (fabsf(v) >= thv) ? v : 0.f;
      }
      P[i] = v;
    }
  };

#if SAN_USE_ASYNC
  if constexpr (!MASKED_ACCUM) {
    // Async global->LDS staging (ASYNCcnt path): valid cells are written by
    // the async unit, padding/OOB cells by the DS pipe -- disjoint addresses,
    // so the two write streams need no mutual ordering.
    for (int i = lane; i < PR * LW; i += 32) {
      const int row = i / LW, col = i % LW;
      const int gy = y0 - R + row, gx = x0 - R + col;
      if (col < PW && gy >= 0 && gy < IMG_H && gx >= 0 && gx < IMG_W) {
        __builtin_amdgcn_global_load_async_to_lds_b32(
            (g_i32*)(src + gy * IMG_W + gx),
            (l_i32*)&P[i],
            /*offset=*/0, /*cpol=*/0);
      } else {
        P[i] = 0.f;
      }
    }
    __builtin_amdgcn_s_wait_asynccnt(0);
  } else {
    stage_sync();
  }
#else
  stage_sync();
#endif
  __syncthreads();

  const int mn   = lane & 15;                 // A-row M == B-col N for this lane
  const int half = lane >> 4;
  const int kb   = half * 2;                  // K pair base held by this lane

  float res[8];

#if SAN_USE_WMMA
  v8f acc = {};
  for (int dy = 0; dy < L; ++dy) {
    const float* prow = &P[(mn + dy) * LW];   // A row for this lane at this dy
    const float* wrow = &wsh[dy * L];
    for (int c = 0; c < NC; ++c) {
      const int t0 = 4 * c + kb;
      v2f a;
      a.x = prow[t0];
      a.y = prow[t0 + 1];
      const int j0 = t0 - mn;                 // band index w[dy][t - n]
      v2f bb;
      bb.x = (j0 >= 0 && j0 < L) ? wrow[j0] : 0.f;
      bb.y = (j0 + 1 >= 0 && j0 + 1 < L) ? wrow[j0 + 1] : 0.f;
      // 8 args: (neg_a, A, neg_b, B, c_mod, C, reuse_a, reuse_b)
      acc = __builtin_amdgcn_wmma_f32_16x16x4_f32(
          false, a, false, bb, (short)0, acc, false, false);
    }
  }
  for (int v = 0; v < 8; ++v) res[v] = acc[v];
#else
  // VALU fallback (same D fragment per lane)
  for (int v = 0; v < 8; ++v) res[v] = 0.f;
  for (int dy = 0; dy < L; ++dy)
    for (int dx = 0; dx < L; ++dx) {
      const float wv = wsh[dy * L + dx];
      for (int v = 0; v < 8; ++v)
        res[v] += wv * P[(v + 8 * half + dy) * LW + (mn + dx)];
    }
#endif

  float* dst = out + (size_t)b * IMG_N;
  const int col  = x0 + mn;
  const int rowb = y0 + half * 8;
  for (int v = 0; v < 8; ++v) {
    const size_t o = (size_t)(rowb + v) * IMG_W + col;
    if (MASKED_ACCUM)
      dst[o] += res[v];
    else
      dst[o] = res[v];
  }
}

// ---------------------------------------------------------------------------
// Exact k-th-largest |value| per sample: 4-pass radix select over the 31
// magnitude bits (8/8/8/7). One 1024-thread workgroup per sample; sim is
// L2-resident so the 4 passes are cheap.
// ---------------------------------------------------------------------------
__global__ __launch_bounds__(1024) void radix_select_kernel(
    const float* __restrict__ sim, float* __restrict__ th, int k) {
  __shared__ unsigned hist[256];
  __shared__ unsigned s_prefix;
  __shared__ int s_krem;

  const int b = blockIdx.x;
  const int tid = threadIdx.x;
  const float* p = sim + (size_t)b * IMG_N;

  if (tid == 0) { s_prefix = 0u; s_krem = k; }

  const int shifts[4] = {23, 15, 7, 0};
  const int nbits[4]  = {8, 8, 8, 7};

  for (int pass = 0; pass < 4; ++pass) {
    if (tid < 256) hist[tid] = 0u;
    __syncthreads();
    const unsigned prefix = s_prefix;
    const int sh  = shifts[pass];
    const int nb  = nbits[pass];
    const unsigned msk = (1u << nb) - 1u;
    const int hs = sh + nb;  // pass0: key>>31 == 0 == prefix, all match

    for (int i = tid; i < IMG_N; i += 1024) {
      const unsigned key = __float_as_uint(p[i]) & 0x7FFFFFFFu;
      if ((key >> hs) == prefix) atomicAdd(&hist[(key >> sh) & msk], 1u);
    }
    __syncthreads();

    if (tid == 0) {
      const int krem = s_krem;
      unsigned cum = 0;
      for (int bin = (int)msk; bin >= 0; --bin) {
        const unsigned c2 = cum + hist[bin];
        if ((int)c2 >= krem) {
          s_krem = krem - (int)cum;
          s_prefix = (prefix << nb) | (unsigned)bin;
          break;
        }
        cum = c2;
      }
    }
    __syncthreads();
  }

  if (tid == 0) th[b] = __uint_as_float(s_prefix);
}

__global__ __launch_bounds__(256) void zero_out_kernel(float4* __restrict__ o) {
  o[(size_t)blockIdx.x * 256 + threadIdx.x] = make_float4(0.f, 0.f, 0.f, 0.f);
}

// ---------------------------------------------------------------------------
extern "C" void kernel_launch(void* const* d_in, const int* in_sizes, int n_in,
                              void* d_out, int out_size, void* d_ws, size_t ws_size,
                              hipStream_t stream) {
  const float* x  = (const float*)d_in[0];  // [64,1,512,512]
  const float* w3 = (const float*)d_in[1];  // [3,3]
  const float* w5 = (const float*)d_in[2];  // [5,5]
  const float* w7 = (const float*)d_in[3];  // [7,7]
  float* out = (float*)d_out;

  float* sim = (float*)d_ws;                      // 64 MB scratch
  float* th  = sim + (size_t)BATCH * IMG_N;       // 64 thresholds

  const int convGrid = (BATCH * 32 * 32) / 4;     // 4 tiles (waves) per block

  // out = 0  (16,777,216 floats = 4,194,304 float4)
  zero_out_kernel<<<16384, 256, 0, stream>>>((float4*)out);

  // ---- 3x3 ----
  conv_wmma_kernel<3, false><<<convGrid, 128, 0, stream>>>(x, w3, nullptr, sim);
  radix_select_kernel<<<BATCH, 1024, 0, stream>>>(sim, th, TOPK);
  conv_wmma_kernel<3, true><<<convGrid, 128, 0, stream>>>(sim, w3, th, out);

  // ---- 5x5 ----
  conv_wmma_kernel<5, false><<<convGrid, 128, 0, stream>>>(x, w5, nullptr, sim);
  radix_select_kernel<<<BATCH, 1024, 0, stream>>>(sim, th, TOPK);
  conv_wmma_kernel<5, true><<<convGrid, 128, 0, stream>>>(sim, w5, th, out);

  // ---- 7x7 ----
  conv_wmma_kernel<7, false><<<convGrid, 128, 0, stream>>>(x, w7, nullptr, sim);
  radix_select_kernel<<<BATCH, 1024, 0, stream>>>(sim, th, TOPK);
  conv_wmma_kernel<7, true><<<convGrid, 128, 0, stream>>>(sim, w7, th, out);
}